// GraphNodeEdgeConvLayer_15126874816758
// MI455X (gfx1250) — compile-verified
//
#include <hip/hip_runtime.h>

// Problem constants (from reference)
constexpr int BATCH = 4;
constexpr int NNODE = 512;
constexpr int DNODE = 128;
constexpr int DEDGE = 64;
constexpr int HDIM  = 128;

typedef float v2f __attribute__((ext_vector_type(2)));
typedef float v8f __attribute__((ext_vector_type(8)));

__device__ __forceinline__ v8f wmma_f32(v2f a, v2f b, v8f c) {
  // V_WMMA_F32_16X16X4_F32 : D = A(16x4) * B(4x16) + C(16x16), fp32 exact
  return __builtin_amdgcn_wmma_f32_16x16x4_f32(
      /*neg_a=*/false, a, /*neg_b=*/false, b,
      /*c_mod=*/(short)0, c, /*reuse_a=*/false, /*reuse_b=*/false);
}

__device__ __forceinline__ v8f vzero8() {
  v8f z = {0.f,0.f,0.f,0.f,0.f,0.f,0.f,0.f};
  return z;
}

// CDNA5 async global->LDS copy (16B per lane), tracked by ASYNCcnt.
// Generic pointer low 32 bits == workgroup-relative LDS byte address.
__device__ __forceinline__ void async_b128(void* lds, const void* g) {
  unsigned l = (unsigned)(size_t)lds;
  asm volatile("global_load_async_to_lds_b128 %0, %1, off"
               :: "v"(l), "v"(g) : "memory");
}
__device__ __forceinline__ void wait_async() {
  asm volatile("s_wait_asynccnt 0x0" ::: "memory");
}

// ---------------------------------------------------------------------------
// Kernel 1: per (b, 16-row i tile):
//   agg_n = adjTile @ node_feat[b]            (WMMA, K=512, LDS via async DMA)
//   edge_agg = sum_j adj[b,i,j]*edge[b,i,j,:] (streaming, NT loads)
//   h_out = relu(relu(agg_n@Wn+bn) + relu(edge_agg@We+be))
//   pi = h_out@Wu[:H], pj = h_out@Wu[H:2H]    (to workspace)
// ---------------------------------------------------------------------------
__global__ __launch_bounds__(256) void k_node(
    const float* __restrict__ node, const float* __restrict__ adj,
    const float* __restrict__ edge, const float* __restrict__ Wn,
    const float* __restrict__ bn,   const float* __restrict__ We,
    const float* __restrict__ be,   const float* __restrict__ Wu,
    float* __restrict__ hout, float* __restrict__ pi, float* __restrict__ pj)
{
  __shared__ __align__(16) float adjS[16 * (NNODE + 4)];   // 33 KB, +4 pad
  __shared__ __align__(16) float nodeS[64 * DNODE];        // 32 KB, K-chunk
  __shared__ __align__(16) float aggS[16 * (DNODE + 4)];
  __shared__ __align__(16) float eaggS[16 * (DEDGE + 4)];
  __shared__ __align__(16) float hS[16 * (HDIM + 4)];

  const int tid  = threadIdx.x;
  const int wave = tid >> 5;
  const int lane = tid & 31;
  const int m    = lane & 15;
  const int hi   = lane >> 4;
  const int n0   = wave << 4;                 // this wave's 16-col output tile
  const int b    = blockIdx.x >> 5;           // 32 i-tiles per batch
  const int i0   = (blockIdx.x & 31) << 4;

  // adj tile 16 x 512 -> LDS via async DMA (row pad 4 floats keeps 16B align:
  // 516*4 = 2064 B per row)
  for (int idx = tid; idx < 16 * (NNODE / 4); idx += 256) {
    int r = idx >> 7, c4 = (idx & 127) << 2;
    async_b128(&adjS[r * (NNODE + 4) + c4],
               &adj[(size_t)(b * NNODE + i0 + r) * NNODE + c4]);
  }

  // ---- agg_n: (16x512)@(512x128), K-chunked, chunks DMA'd to LDS
  v8f accA = vzero8();
  for (int kb = 0; kb < NNODE; kb += 64) {
    for (int idx = tid; idx < (64 * DNODE) / 4; idx += 256)
      async_b128(&nodeS[idx << 2],
                 &node[(size_t)(b * NNODE + kb) * DNODE + (idx << 2)]);
    wait_async();           // first iter also covers adjS
    __syncthreads();
    #pragma unroll
    for (int k = 0; k < 64; k += 4) {
      v2f a, w;
      a.x = adjS[m * (NNODE + 4) + kb + k + 2 * hi];
      a.y = adjS[m * (NNODE + 4) + kb + k + 2 * hi + 1];
      w.x = nodeS[(k + 2 * hi) * DNODE + n0 + m];
      w.y = nodeS[(k + 2 * hi + 1) * DNODE + n0 + m];
      accA = wmma_f32(a, w, accA);
    }
    __syncthreads();
  }
  #pragma unroll
  for (int r = 0; r < 8; ++r)
    aggS[(r + 8 * hi) * (DNODE + 4) + n0 + m] = accA[r];

  // ---- edge_agg: thread -> column c, rows g,g+4,g+8,g+12 (row uniform per
  //      wave -> LDS broadcast of adj; lanes consecutive in c -> coalesced,
  //      non-temporal streaming reads of edge_feat)
  {
    const int c = tid & (DEDGE - 1);
    const int g = tid >> 6;
    float s0 = 0.f, s1 = 0.f, s2 = 0.f, s3 = 0.f;
    for (int j = 0; j < NNODE; ++j) {
      float a0 = adjS[(g)      * (NNODE + 4) + j];
      float a1 = adjS[(g + 4)  * (NNODE + 4) + j];
      float a2 = adjS[(g + 8)  * (NNODE + 4) + j];
      float a3 = adjS[(g + 12) * (NNODE + 4) + j];
      s0 += a0 * __builtin_nontemporal_load(
          &edge[((size_t)(b * NNODE + i0 + g)      * NNODE + j) * DEDGE + c]);
      s1 += a1 * __builtin_nontemporal_load(
          &edge[((size_t)(b * NNODE + i0 + g + 4)  * NNODE + j) * DEDGE + c]);
      s2 += a2 * __builtin_nontemporal_load(
          &edge[((size_t)(b * NNODE + i0 + g + 8)  * NNODE + j) * DEDGE + c]);
      s3 += a3 * __builtin_nontemporal_load(
          &edge[((size_t)(b * NNODE + i0 + g + 12) * NNODE + j) * DEDGE + c]);
    }
    eaggS[(g)      * (DEDGE + 4) + c] = s0;
    eaggS[(g + 4)  * (DEDGE + 4) + c] = s1;
    eaggS[(g + 8)  * (DEDGE + 4) + c] = s2;
    eaggS[(g + 12) * (DEDGE + 4) + c] = s3;
  }
  __syncthreads();

  // ---- h_out = relu(relu(agg@Wn+bn) + relu(eagg@We+be))
  v8f accN = vzero8(), accE = vzero8();
  #pragma unroll
  for (int k = 0; k < DNODE; k += 4) {
    v2f a, w;
    a.x = aggS[m * (DNODE + 4) + k + 2 * hi];
    a.y = aggS[m * (DNODE + 4) + k + 2 * hi + 1];
    w.x = Wn[(k + 2 * hi) * HDIM + n0 + m];
    w.y = Wn[(k + 2 * hi + 1) * HDIM + n0 + m];
    accN = wmma_f32(a, w, accN);
  }
  #pragma unroll
  for (int k = 0; k < DEDGE; k += 4) {
    v2f a, w;
    a.x = eaggS[m * (DEDGE + 4) + k + 2 * hi];
    a.y = eaggS[m * (DEDGE + 4) + k + 2 * hi + 1];
    w.x = We[(k + 2 * hi) * HDIM + n0 + m];
    w.y = We[(k + 2 * hi + 1) * HDIM + n0 + m];
    accE = wmma_f32(a, w, accE);
  }
  const float bnv = bn[n0 + m], bev = be[n0 + m];
  #pragma unroll
  for (int r = 0; r < 8; ++r) {
    float hn = fmaxf(accN[r] + bnv, 0.f);
    float he = fmaxf(accE[r] + bev, 0.f);
    float h  = fmaxf(hn + he, 0.f);
    int row = r + 8 * hi;
    hS[row * (HDIM + 4) + n0 + m] = h;
    hout[(size_t)(b * NNODE + i0 + row) * HDIM + n0 + m] = h;
  }
  __syncthreads();

  // ---- pi = h@Wu[:H], pj = h@Wu[H:2H]  (shared A operand, two B streams)
  v8f accP = vzero8(), accQ = vzero8();
  #pragma unroll
  for (int k = 0; k < HDIM; k += 4) {
    v2f a, w0, w1;
    a.x  = hS[m * (HDIM + 4) + k + 2 * hi];
    a.y  = hS[m * (HDIM + 4) + k + 2 * hi + 1];
    w0.x = Wu[(k + 2 * hi) * HDIM + n0 + m];
    w0.y = Wu[(k + 2 * hi + 1) * HDIM + n0 + m];
    w1.x = Wu[(HDIM + k + 2 * hi) * HDIM + n0 + m];
    w1.y = Wu[(HDIM + k + 2 * hi + 1) * HDIM + n0 + m];
    accP = wmma_f32(a, w0, accP);
    accQ = wmma_f32(a, w1, accQ);
  }
  #pragma unroll
  for (int r = 0; r < 8; ++r) {
    int row = r + 8 * hi;
    size_t o = (size_t)(b * NNODE + i0 + row) * HDIM + n0 + m;
    pi[o] = accP[r];
    pj[o] = accQ[r];
  }
}

// ---------------------------------------------------------------------------
// Kernel 2: one WG per (b,i); We/Wu2/pi+bu resident in LDS (async preload);
// j streamed in 32-row tiles with DOUBLE-BUFFERED async X fetch:
//   iter: issue DMA for tile jb+32 -> XS[buf^1]
//         stage1: T1 = relu(XS[buf]@We+be)      (WMMA) -> LDS
//         stage2: T2 = T1@Wu2                   (WMMA)
//         relu(T2 + pi + pj + bu)*adj -> e_out  (non-temporal stores)
//         s_wait_asynccnt 0 ; barrier ; buf ^= 1
// e_proj never touches HBM; DMA overlaps both WMMA stages.
// ---------------------------------------------------------------------------
__global__ __launch_bounds__(256) void k_edge(
    const float* __restrict__ adj, const float* __restrict__ edge,
    const float* __restrict__ We,  const float* __restrict__ be,
    const float* __restrict__ Wu,  const float* __restrict__ bu,
    const float* __restrict__ pi,  const float* __restrict__ pj,
    float* __restrict__ eout)
{
  __shared__ __align__(16) float WeS[DEDGE * HDIM];        // 32 KB
  __shared__ __align__(16) float WuS[HDIM * HDIM];         // 64 KB (Wu[2H:])
  __shared__ __align__(16) float XS[2][32 * (DEDGE + 4)];  // 2 x 8.5 KB
  __shared__ __align__(16) float T1S[32 * (HDIM + 4)];     // 16.5 KB
  __shared__ float pibS[HDIM];
  __shared__ float beS[HDIM];
  __shared__ float adjS[32];

  const int tid  = threadIdx.x;
  const int wave = tid >> 5;
  const int lane = tid & 31;
  const int m    = lane & 15;
  const int hi   = lane >> 4;
  const int n0   = wave << 4;
  const int b    = blockIdx.x >> 9;           // / NNODE
  const int i    = blockIdx.x & (NNODE - 1);

  // async preload of resident weights (contiguous 16B chunks)
  for (int idx = tid; idx < (DEDGE * HDIM) / 4; idx += 256)
    async_b128(&WeS[idx << 2], &We[idx << 2]);
  for (int idx = tid; idx < (HDIM * HDIM) / 4; idx += 256)
    async_b128(&WuS[idx << 2], &Wu[2 * HDIM * HDIM + (idx << 2)]);

  const size_t rowbase = (size_t)(b * NNODE + i) * NNODE;  // (b,i,*) pair index

  // preload X tile 0 into XS[0] (row pad 4 floats: 68*4=272 B, 16B aligned)
  for (int idx = tid; idx < 32 * (DEDGE / 4); idx += 256) {
    int r = idx >> 4, c4 = (idx & 15) << 2;
    async_b128(&XS[0][r * (DEDGE + 4) + c4],
               &edge[(rowbase + r) * DEDGE + c4]);
  }
  if (tid < HDIM) {
    pibS[tid] = pi[(size_t)(b * NNODE + i) * HDIM + tid] + bu[tid];
    beS[tid]  = be[tid];
  }
  wait_async();
  __syncthreads();

  int buf = 0;
  for (int jb = 0; jb < NNODE; jb += 32) {
    if (tid < 32) adjS[tid] = adj[rowbase + jb + tid];
    // kick off DMA for the next tile into the other buffer (overlaps stages)
    if (jb + 32 < NNODE) {
      for (int idx = tid; idx < 32 * (DEDGE / 4); idx += 256) {
        int r = idx >> 4, c4 = (idx & 15) << 2;
        async_b128(&XS[buf ^ 1][r * (DEDGE + 4) + c4],
                   &edge[(rowbase + jb + 32 + r) * DEDGE + c4]);
      }
    }

    // stage 1: T1(32x128) = relu(X @ We + be); wave: 16 cols x 2 row-tiles
    v8f acc0 = vzero8(), acc1 = vzero8();
    #pragma unroll
    for (int k = 0; k < DEDGE; k += 4) {
      v2f w, a0, a1;
      w.x  = WeS[(k + 2 * hi) * HDIM + n0 + m];
      w.y  = WeS[(k + 2 * hi + 1) * HDIM + n0 + m];
      a0.x = XS[buf][m * (DEDGE + 4) + k + 2 * hi];
      a0.y = XS[buf][m * (DEDGE + 4) + k + 2 * hi + 1];
      a1.x = XS[buf][(m + 16) * (DEDGE + 4) + k + 2 * hi];
      a1.y = XS[buf][(m + 16) * (DEDGE + 4) + k + 2 * hi + 1];
      acc0 = wmma_f32(a0, w, acc0);
      acc1 = wmma_f32(a1, w, acc1);
    }
    {
      const float bev = beS[n0 + m];
      #pragma unroll
      for (int r = 0; r < 8; ++r) {
        T1S[(r + 8 * hi) * (HDIM + 4) + n0 + m]      = fmaxf(acc0[r] + bev, 0.f);
        T1S[(r + 8 * hi + 16) * (HDIM + 4) + n0 + m] = fmaxf(acc1[r] + bev, 0.f);
      }
    }
    __syncthreads();

    // stage 2: T2 = T1 @ Wu2 ; epilogue: relu(T2 + pi + pj + bu) * adj
    acc0 = vzero8(); acc1 = vzero8();
    #pragma unroll
    for (int k = 0; k < HDIM; k += 4) {
      v2f w, a0, a1;
      w.x  = WuS[(k + 2 * hi) * HDIM + n0 + m];
      w.y  = WuS[(k + 2 * hi + 1) * HDIM + n0 + m];
      a0.x = T1S[m * (HDIM + 4) + k + 2 * hi];
      a0.y = T1S[m * (HDIM + 4) + k + 2 * hi + 1];
      a1.x = T1S[(m + 16) * (HDIM + 4) + k + 2 * hi];
      a1.y = T1S[(m + 16) * (HDIM + 4) + k + 2 * hi + 1];
      acc0 = wmma_f32(a0, w, acc0);
      acc1 = wmma_f32(a1, w, acc1);
    }
    {
      const float pib = pibS[n0 + m];
      #pragma unroll
      for (int r = 0; r < 8; ++r) {
        int jl0 = r + 8 * hi;
        int jl1 = jl0 + 16;
        float pj0 = pj[(size_t)(b * NNODE + jb + jl0) * HDIM + n0 + m];
        float pj1 = pj[(size_t)(b * NNODE + jb + jl1) * HDIM + n0 + m];
        // write-once 537 MB stream: keep it out of the caches
        __builtin_nontemporal_store(
            fmaxf(acc0[r] + pib + pj0, 0.f) * adjS[jl0],
            &eout[(rowbase + jb + jl0) * HDIM + n0 + m]);
        __builtin_nontemporal_store(
            fmaxf(acc1[r] + pib + pj1, 0.f) * adjS[jl1],
            &eout[(rowbase + jb + jl1) * HDIM + n0 + m]);
      }
    }
    wait_async();        // next tile's DMA has had 96 WMMAs to complete
    __syncthreads();
    buf ^= 1;
  }
}

extern "C" void kernel_launch(void* const* d_in, const int* in_sizes, int n_in,
                              void* d_out, int out_size, void* d_ws, size_t ws_size,
                              hipStream_t stream) {
  const float* node = (const float*)d_in[0];
  const float* adj  = (const float*)d_in[1];
  const float* edge = (const float*)d_in[2];
  const float* Wn   = (const float*)d_in[3];
  const float* bn   = (const float*)d_in[4];
  const float* We   = (const float*)d_in[5];
  const float* be   = (const float*)d_in[6];
  const float* Wu   = (const float*)d_in[7];
  const float* bu   = (const float*)d_in[8];

  float* hout = (float*)d_out;                                  // [B,N,H]
  float* eout = hout + (size_t)BATCH * NNODE * HDIM;            // [B,N,N,H]
  float* pi   = (float*)d_ws;                                   // [B,N,H]
  float* pj   = pi + (size_t)BATCH * NNODE * HDIM;              // [B,N,H]

  k_node<<<BATCH * (NNODE / 16), 256, 0, stream>>>(
      node, adj, edge, Wn, bn, We, be, Wu, hout, pi, pj);
  k_edge<<<BATCH * NNODE, 256, 0, stream>>>(
      adj, edge, We, be, Wu, bu, pi, pj, eout);
}